// EncoderRNN_18863496364259
// MI455X (gfx1250) — compile-verified
//
#include <hip/hip_runtime.h>
#include <cmath>

// ---------------------------------------------------------------------------
// 2-layer GRU single step (batch=1, seq=1), fp32.
//   x  = emb[input_id]                       [1024]
//   h1 = GRUCell(x,  h1; w_ih1, w_hh1, ...)  [1024]
//   h2 = GRUCell(h1, h2; w_ih2, w_hh2, ...)  [512]
// Memory-bound (~34.6 MB weights; fits in 192 MB L2 across replays). GEMVs
// run through V_WMMA_F32_16X16X4_F32: A = 16x4 weight tile, B = x-chunk
// broadcast into all 16 columns, so D's column 0 carries the 16 row dots.
// Split-K x8 (one wave per slice) for memory-level parallelism; partials are
// reduced in a fixed order by a combine kernel (deterministic, no atomics).
// ---------------------------------------------------------------------------

typedef __attribute__((ext_vector_type(2))) float v2f;
typedef __attribute__((ext_vector_type(8))) float v8f;

#define H1 1024
#define H2 512
#define NR1 (6 * H1)   // 6144 partial rows for layer 1 (gx:3H1 then gh:3H1)
#define NR2 (6 * H2)   // 3072 partial rows for layer 2
#define NSLICE 8       // split-K factor (== waves per block)

// Inner split-K slice: KW is compile-time so the loop unrolls x8 into
// clause-batched global_load_b64 streams feeding chained WMMAs.
template <int KW>
__device__ __forceinline__ v8f dot_slice(const float* __restrict__ wrow,
                                         const float* __restrict__ xrow,
                                         int k0)
{
    v8f acc = {0.f, 0.f, 0.f, 0.f, 0.f, 0.f, 0.f, 0.f};
    #pragma unroll 8
    for (int kk = 0; kk < KW; kk += 4) {
        v2f a = *(const v2f*)(wrow + k0 + kk);   // W[row, k + koff .. +1]
        v2f b = *(const v2f*)(xrow + k0 + kk);   // x[k + koff .. +1] (bcast)
        acc = __builtin_amdgcn_wmma_f32_16x16x4_f32(
                  false, a, false, b, (short)0, acc, false, false);
    }
    return acc;
}

__device__ __forceinline__ void store_partial(float* __restrict__ part,
                                              int NR, int slice, int prow,
                                              int lane, const v8f& acc)
{
    // Column N=0 lives in lanes 0 (M=0..7) and 16 (M=8..15), VGPRs 0..7.
    if ((lane & 15) == 0) {
        float* dst = part + (size_t)slice * NR + prow + 8 * (lane >> 4);
        #pragma unroll
        for (int r = 0; r < 8; ++r) dst[r] = acc[r];
    }
}

// ---- layer-1 GEMV: tiles 0..191 -> w_ih1 (vec = emb row), 192..383 -> w_hh1
__global__ __launch_bounds__(256)
void gru_gemv_l1(const float* __restrict__ w_ih, const float* __restrict__ w_hh,
                 const float* __restrict__ emb,  const int*   __restrict__ ids,
                 const float* __restrict__ h1prev, float* __restrict__ part)
{
    __shared__ float svec[H1];
    const int  tile = blockIdx.x;                 // 0..383
    const bool ih   = tile < 192;
    const float* W   = ih ? w_ih : w_hh;
    const float* vec = ih ? (emb + (size_t)ids[0] * H1) : h1prev;
    const int rowBase = (ih ? tile : tile - 192) * 16;   // row in W
    const int prow    = tile * 16;                       // row in partial buf

    for (int i = threadIdx.x; i < H1; i += blockDim.x) svec[i] = vec[i];
    __syncthreads();

    const int lane  = threadIdx.x & 31;   // wave32
    const int slice = threadIdx.x >> 5;   // 0..7 (K slice)
    const int koff  = 2 * (lane >> 4);    // half 0: K0/K1, half 1: K2/K3

    const float* wrow = W + (size_t)(rowBase + (lane & 15)) * H1 + koff;
    const float* xrow = svec + koff;

    const v8f acc = dot_slice<H1 / NSLICE>(wrow, xrow, slice * (H1 / NSLICE));
    store_partial(part, NR1, slice, prow, lane, acc);
}

// ---- layer-2 GEMV: tiles 0..95 -> w_ih2 (K=1024, vec=h1), 96..191 -> w_hh2 (K=512)
__global__ __launch_bounds__(256)
void gru_gemv_l2(const float* __restrict__ w_ih, const float* __restrict__ w_hh,
                 const float* __restrict__ h1,   const float* __restrict__ h2prev,
                 float* __restrict__ part)
{
    __shared__ float svec[H1];
    const int  tile = blockIdx.x;                 // 0..191
    const bool ih   = tile < 96;
    const float* W   = ih ? w_ih : w_hh;
    const float* vec = ih ? h1 : h2prev;
    const int K       = ih ? H1 : H2;             // also leading dimension
    const int rowBase = (ih ? tile : tile - 96) * 16;
    const int prow    = tile * 16;

    for (int i = threadIdx.x; i < K; i += blockDim.x) svec[i] = vec[i];
    __syncthreads();

    const int lane  = threadIdx.x & 31;
    const int slice = threadIdx.x >> 5;
    const int koff  = 2 * (lane >> 4);

    const float* wrow = W + (size_t)(rowBase + (lane & 15)) * K + koff;
    const float* xrow = svec + koff;

    // Block-uniform branch (EXEC stays all-1s); both sides get compile-time
    // trip counts so the x8 unroll / load clauses materialize.
    if (ih) {
        const v8f acc = dot_slice<H1 / NSLICE>(wrow, xrow, slice * (H1 / NSLICE));
        store_partial(part, NR2, slice, prow, lane, acc);
    } else {
        const v8f acc = dot_slice<H2 / NSLICE>(wrow, xrow, slice * (H2 / NSLICE));
        store_partial(part, NR2, slice, prow, lane, acc);
    }
}

__device__ __forceinline__ float sigm(float v) { return 1.f / (1.f + __expf(-v)); }

// ---- combine layer 1: fixed-order slice reduction + gate math -> h1
__global__ __launch_bounds__(256)
void gru_combine_l1(const float* __restrict__ part,
                    const float* __restrict__ b_ih, const float* __restrict__ b_hh,
                    const float* __restrict__ h1prev, float* __restrict__ h1out)
{
    const int m = blockIdx.x * blockDim.x + threadIdx.x;   // 0..1023
    float xr = b_ih[m], xz = b_ih[H1 + m], xn = b_ih[2 * H1 + m];
    float hr = b_hh[m], hz = b_hh[H1 + m], hn = b_hh[2 * H1 + m];
    #pragma unroll
    for (int s = 0; s < NSLICE; ++s) {
        const float* p = part + (size_t)s * NR1;
        xr += p[m];            xz += p[H1 + m];     xn += p[2 * H1 + m];
        hr += p[3 * H1 + m];   hz += p[4 * H1 + m]; hn += p[5 * H1 + m];
    }
    const float r = sigm(xr + hr);
    const float z = sigm(xz + hz);
    const float n = tanhf(xn + r * hn);
    h1out[m] = (1.f - z) * n + z * h1prev[m];
}

// ---- combine layer 2 -> h2, written to both output slots
__global__ __launch_bounds__(256)
void gru_combine_l2(const float* __restrict__ part,
                    const float* __restrict__ b_ih, const float* __restrict__ b_hh,
                    const float* __restrict__ h2prev,
                    float* __restrict__ out0, float* __restrict__ out2)
{
    const int m = blockIdx.x * blockDim.x + threadIdx.x;   // 0..511
    float xr = b_ih[m], xz = b_ih[H2 + m], xn = b_ih[2 * H2 + m];
    float hr = b_hh[m], hz = b_hh[H2 + m], hn = b_hh[2 * H2 + m];
    #pragma unroll
    for (int s = 0; s < NSLICE; ++s) {
        const float* p = part + (size_t)s * NR2;
        xr += p[m];            xz += p[H2 + m];     xn += p[2 * H2 + m];
        hr += p[3 * H2 + m];   hz += p[4 * H2 + m]; hn += p[5 * H2 + m];
    }
    const float r = sigm(xr + hr);
    const float z = sigm(xz + hz);
    const float n = tanhf(xn + r * hn);
    const float h2 = (1.f - z) * n + z * h2prev[m];
    out0[m] = h2;   // "output"
    out2[m] = h2;   // new hidden_2
}

extern "C" void kernel_launch(void* const* d_in, const int* in_sizes, int n_in,
                              void* d_out, int out_size, void* d_ws, size_t ws_size,
                              hipStream_t stream)
{
    (void)in_sizes; (void)n_in; (void)out_size; (void)ws_size;

    const int*   ids   = (const int*)  d_in[0];
    const float* h1p   = (const float*)d_in[1];
    const float* h2p   = (const float*)d_in[2];
    const float* emb   = (const float*)d_in[3];
    const float* w_ih1 = (const float*)d_in[4];
    const float* w_hh1 = (const float*)d_in[5];
    const float* b_ih1 = (const float*)d_in[6];
    const float* b_hh1 = (const float*)d_in[7];
    const float* w_ih2 = (const float*)d_in[8];
    const float* w_hh2 = (const float*)d_in[9];
    const float* b_ih2 = (const float*)d_in[10];
    const float* b_hh2 = (const float*)d_in[11];

    float* out   = (float*)d_out;           // [h2(512) | h1(1024) | h2(512)]
    float* part1 = (float*)d_ws;            // 8*6144 floats
    float* part2 = part1 + NSLICE * NR1;    // 8*3072 floats (288 KB total)
    float* h1out = out + H2;                // d_out[512 .. 1536)

    gru_gemv_l1   <<<384,      256, 0, stream>>>(w_ih1, w_hh1, emb, ids, h1p, part1);
    gru_combine_l1<<<H1 / 256, 256, 0, stream>>>(part1, b_ih1, b_hh1, h1p, h1out);
    gru_gemv_l2   <<<192,      256, 0, stream>>>(w_ih2, w_hh2, h1out, h2p, part2);
    gru_combine_l2<<<H2 / 256, 256, 0, stream>>>(part2, b_ih2, b_hh2, h2p,
                                                 out, out + 3 * H2);
}